// MultiplicativeAttention_59811714564576
// MI455X (gfx1250) — compile-verified
//
#include <hip/hip_runtime.h>
#include <hip/hip_bf16.h>
#include <math.h>

// ---------------------------------------------------------------------------
// MultiplicativeAttention, rewritten via associativity:
//   scores[b,t] = (context[b,t,:] @ W) . query[b,:] = context[b,t,:] . (W @ query[b,:])
// so the big GEMM disappears; the kernel becomes one streaming pass over
// context (268 MB -> ~11.5us at 23.3 TB/s).  W @ query is done with
// V_WMMA_F32_16X16X4_F32 (fp32 WMMA, precision-compatible with reference).
// ---------------------------------------------------------------------------

#define B_   64
#define T_   2048
#define D_   512
#define TC_  128                 // rows of context per block in the stream pass
#define NCHUNK_ (T_ / TC_)       // 16

typedef __attribute__((ext_vector_type(2))) float v2f;
typedef __attribute__((ext_vector_type(4))) float v4f;
typedef __attribute__((ext_vector_type(8))) float v8f;

// ---------------------------------------------------------------------------
// K1: wq[b,d] = sum_e W[d,e] * query[b,e]     (q[64,512] @ W^T[512,512])
// One wave per 16x16 output tile; K-loop of 128 chained V_WMMA_F32_16X16X4_F32.
// A (16x4, M=b rows)  : lanes 0-15 M=lane {K=k0,k0+1}; lanes 16-31 M=lane-16 {K=k0+2,k0+3}
// B (4x16, N=d cols)  : lanes 0-15 N=lane {K=k0,k0+1}; lanes 16-31 N=lane-16 {K=k0+2,k0+3}
// C/D (16x16 f32)     : VGPR r: lanes 0-15 -> (M=r, N=lane); lanes 16-31 -> (M=8+r, N=lane-16)
// ---------------------------------------------------------------------------
__global__ __launch_bounds__(32) void wq_wmma_kernel(
    const float* __restrict__ W,     // [D_, D_]  (Dc x Dq)
    const float* __restrict__ query, // [B_, D_]
    float* __restrict__ wq)          // [B_, D_]
{
    const int tile = blockIdx.x;          // 0..127
    const int bt = tile >> 5;             // 4 b-tiles
    const int dt = tile & 31;             // 32 d-tiles
    const int b0 = bt * 16;
    const int d0 = dt * 16;
    const int lane = threadIdx.x;
    const int half = lane >> 4;           // 0 or 1
    const int l    = lane & 15;

    const float* qrow = query + (size_t)(b0 + l) * D_;  // A: M = b0+l
    const float* wrow = W     + (size_t)(d0 + l) * D_;  // B: N = d0+l

    v8f c = {};
    for (int k0 = 0; k0 < D_; k0 += 4) {
        const int ka = k0 + half * 2;
        v2f a, bm;
        a.x  = qrow[ka];
        a.y  = qrow[ka + 1];
        bm.x = wrow[ka];
        bm.y = wrow[ka + 1];
        c = __builtin_amdgcn_wmma_f32_16x16x4_f32(
                /*neg_a=*/false, a, /*neg_b=*/false, bm,
                /*c_mod=*/(short)0, c, /*reuse_a=*/false, /*reuse_b=*/false);
    }

    // D layout -> wq
#pragma unroll
    for (int r = 0; r < 8; ++r) {
        wq[(size_t)(b0 + half * 8 + r) * D_ + d0 + l] = c[r];
    }
}

// ---------------------------------------------------------------------------
// K2: single streaming pass over context. One block = (b, chunk of 128 rows).
// Wave w handles rows t0+w, t0+w+8, ... (16 rows). Lane holds 16 d-columns
// laid out as d = k*128 + lane*4 + j (k=0..3, j=0..3) so every
// global_load_b128 is perfectly coalesced (wave covers 512B contiguous).
// Per row: s = c.wq (butterfly reduce), w = exp(tanh(s))*mask, acc += w*c.
// Per-wave partials reduced through LDS; chunk partial -> workspace.
// ---------------------------------------------------------------------------
__global__ __launch_bounds__(256) void attn_stream_kernel(
    const float* __restrict__ ctx,           // [B_, T_, D_]
    const unsigned char* __restrict__ mask,  // [B_, T_] (bool, 1 byte)
    const float* __restrict__ wq,            // [B_, D_]
    float* __restrict__ accPart,             // [B_*NCHUNK_, D_]
    float* __restrict__ zPart)               // [B_*NCHUNK_]
{
    const int blk   = blockIdx.x;
    const int b     = blk / NCHUNK_;
    const int chunk = blk - b * NCHUNK_;
    const int t0    = chunk * TC_;
    const int tid   = threadIdx.x;
    const int wave  = tid >> 5;
    const int lane  = tid & 31;

    __shared__ float s_acc[8 * D_];
    __shared__ float s_z[8];

    // this lane's 16 columns of wq
    const float* wqb = wq + (size_t)b * D_;
    const int dbase = lane * 4;
    v4f wqv[4];
#pragma unroll
    for (int k = 0; k < 4; ++k)
        wqv[k] = *(const v4f*)(wqb + k * 128 + dbase);

    v4f acc0 = (v4f)(0.f), acc1 = (v4f)(0.f), acc2 = (v4f)(0.f), acc3 = (v4f)(0.f);
    float z = 0.f;

    const float* cb = ctx + ((size_t)b * T_ + t0) * D_;
    const unsigned char* mb = mask + (size_t)b * T_ + t0;

    for (int t = wave; t < TC_; t += 8) {
        const float* row = cb + (size_t)t * D_;
        // streamed once -> non-temporal loads (don't pollute 192MB L2)
        v4f c0 = __builtin_nontemporal_load((const v4f*)(row + 0 * 128 + dbase));
        v4f c1 = __builtin_nontemporal_load((const v4f*)(row + 1 * 128 + dbase));
        v4f c2 = __builtin_nontemporal_load((const v4f*)(row + 2 * 128 + dbase));
        v4f c3 = __builtin_nontemporal_load((const v4f*)(row + 3 * 128 + dbase));

        float p = c0.x * wqv[0].x;
        p = fmaf(c0.y, wqv[0].y, p); p = fmaf(c0.z, wqv[0].z, p); p = fmaf(c0.w, wqv[0].w, p);
        p = fmaf(c1.x, wqv[1].x, p); p = fmaf(c1.y, wqv[1].y, p);
        p = fmaf(c1.z, wqv[1].z, p); p = fmaf(c1.w, wqv[1].w, p);
        p = fmaf(c2.x, wqv[2].x, p); p = fmaf(c2.y, wqv[2].y, p);
        p = fmaf(c2.z, wqv[2].z, p); p = fmaf(c2.w, wqv[2].w, p);
        p = fmaf(c3.x, wqv[3].x, p); p = fmaf(c3.y, wqv[3].y, p);
        p = fmaf(c3.z, wqv[3].z, p); p = fmaf(c3.w, wqv[3].w, p);

        // wave32 butterfly reduction -> all lanes hold full dot product
        p += __shfl_xor(p, 16);
        p += __shfl_xor(p, 8);
        p += __shfl_xor(p, 4);
        p += __shfl_xor(p, 2);
        p += __shfl_xor(p, 1);

        float w = expf(tanhf(p));
        w = (mb[t] != 0) ? w : 0.f;
        z += w;

        acc0.x = fmaf(w, c0.x, acc0.x); acc0.y = fmaf(w, c0.y, acc0.y);
        acc0.z = fmaf(w, c0.z, acc0.z); acc0.w = fmaf(w, c0.w, acc0.w);
        acc1.x = fmaf(w, c1.x, acc1.x); acc1.y = fmaf(w, c1.y, acc1.y);
        acc1.z = fmaf(w, c1.z, acc1.z); acc1.w = fmaf(w, c1.w, acc1.w);
        acc2.x = fmaf(w, c2.x, acc2.x); acc2.y = fmaf(w, c2.y, acc2.y);
        acc2.z = fmaf(w, c2.z, acc2.z); acc2.w = fmaf(w, c2.w, acc2.w);
        acc3.x = fmaf(w, c3.x, acc3.x); acc3.y = fmaf(w, c3.y, acc3.y);
        acc3.z = fmaf(w, c3.z, acc3.z); acc3.w = fmaf(w, c3.w, acc3.w);
    }

    // stash per-wave partials in LDS
    float* sa = s_acc + wave * D_;
    sa[0 * 128 + dbase + 0] = acc0.x; sa[0 * 128 + dbase + 1] = acc0.y;
    sa[0 * 128 + dbase + 2] = acc0.z; sa[0 * 128 + dbase + 3] = acc0.w;
    sa[1 * 128 + dbase + 0] = acc1.x; sa[1 * 128 + dbase + 1] = acc1.y;
    sa[1 * 128 + dbase + 2] = acc1.z; sa[1 * 128 + dbase + 3] = acc1.w;
    sa[2 * 128 + dbase + 0] = acc2.x; sa[2 * 128 + dbase + 1] = acc2.y;
    sa[2 * 128 + dbase + 2] = acc2.z; sa[2 * 128 + dbase + 3] = acc2.w;
    sa[3 * 128 + dbase + 0] = acc3.x; sa[3 * 128 + dbase + 1] = acc3.y;
    sa[3 * 128 + dbase + 2] = acc3.z; sa[3 * 128 + dbase + 3] = acc3.w;
    if (lane == 0) s_z[wave] = z;   // identical across lanes after butterfly
    __syncthreads();

    if (tid == 0) {
        float zs = 0.f;
#pragma unroll
        for (int w = 0; w < 8; ++w) zs += s_z[w];
        zPart[blk] = zs;
    }
#pragma unroll
    for (int d = tid; d < D_; d += 256) {
        float s = 0.f;
#pragma unroll
        for (int w = 0; w < 8; ++w) s += s_acc[w * D_ + d];
        accPart[(size_t)blk * D_ + d] = s;
    }
}

// ---------------------------------------------------------------------------
// K3: deterministic reduction over the 16 chunks, normalize by (Z + eps).
// ---------------------------------------------------------------------------
__global__ __launch_bounds__(256) void attn_reduce_kernel(
    const float* __restrict__ accPart, const float* __restrict__ zPart,
    float* __restrict__ out)
{
    const int b = blockIdx.x;
    const int tid = threadIdx.x;

    float z = 0.f;
#pragma unroll
    for (int c = 0; c < NCHUNK_; ++c) z += zPart[b * NCHUNK_ + c];
    const float inv = 1.0f / (z + 1e-7f);

#pragma unroll
    for (int d = tid; d < D_; d += 256) {
        float s = 0.f;
#pragma unroll
        for (int c = 0; c < NCHUNK_; ++c)
            s += accPart[(size_t)(b * NCHUNK_ + c) * D_ + d];
        out[(size_t)b * D_ + d] = s * inv;
    }
}

// ---------------------------------------------------------------------------
extern "C" void kernel_launch(void* const* d_in, const int* in_sizes, int n_in,
                              void* d_out, int out_size, void* d_ws, size_t ws_size,
                              hipStream_t stream) {
    const float* ctx  = (const float*)d_in[0];           // [64,2048,512] f32
    const float* qry  = (const float*)d_in[1];           // [64,512]      f32
    const float* W    = (const float*)d_in[2];           // [512,512]     f32
    const unsigned char* mask = (const unsigned char*)d_in[3]; // [64,2048] bool
    float* out = (float*)d_out;                          // [64,512]      f32

    float* ws = (float*)d_ws;
    float* wq      = ws;                                  // 64*512
    float* accPart = wq + B_ * D_;                        // 64*16*512
    float* zPart   = accPart + (size_t)B_ * NCHUNK_ * D_; // 64*16

    // K1: wq = query @ W^T via fp32 WMMA (128 one-wave tiles)
    wq_wmma_kernel<<<dim3(128), dim3(32), 0, stream>>>(W, qry, wq);

    // K2: single streaming pass over context (268 MB, the roofline cost)
    attn_stream_kernel<<<dim3(B_ * NCHUNK_), dim3(256), 0, stream>>>(
        ctx, mask, wq, accPart, zPart);

    // K3: final chunk reduction + normalization
    attn_reduce_kernel<<<dim3(B_), dim3(256), 0, stream>>>(accPart, zPart, out);
}